// NeuralPCFG_29678224015497
// MI455X (gfx1250) — compile-verified
//
#include <hip/hip_runtime.h>
#include <hip/hip_bf16.h>
#include <math.h>

// ---------------------------------------------------------------------------
// Neural PCFG inside algorithm for MI455X (gfx1250, wave32, WMMA f32 16x16x4)
// ---------------------------------------------------------------------------

typedef __attribute__((ext_vector_type(2))) float v2f;
typedef __attribute__((ext_vector_type(8))) float v8f;

namespace {
constexpr int B_  = 16;     // batch
constexpr int N_  = 24;     // sequence length
constexpr int V_  = 10000;  // vocab
constexpr int D_  = 256;    // hidden dim
constexpr int T_  = 40;     // terminals
constexpr int NT_ = 20;     // non-terminals
constexpr int S_  = 60;     // NT + T
constexpr int SP_ = 64;     // padded state stride in chart
constexpr int SS_ = S_ * S_;       // 3600
constexpr int AR_ = NT_ * S_;      // 1200 (stage-1 GEMM N dim)
constexpr int KP_ = 64;            // padded K for inside GEMM (S_ -> 64)
constexpr int RMAX_ = 2304;        // max rows per width: 16*(25-13)*(13-1)
constexpr float NEGF = -1e9f;
}

// ------------------------------ utility ------------------------------------

__global__ void fill_f32(float* __restrict__ p, float v, size_t count) {
    size_t i = (size_t)blockIdx.x * blockDim.x + threadIdx.x;
    if (i < count) p[i] = v;
}

// -------------------- WMMA GEMM: C = A * B^T (+bias, relu, resid) ----------
// A: [M,K] row-major (lda=K). Bt: [N,K] row-major (ldb=K). C: [M,N] (ldc=N).
// One wave per 16x16 tile.  K must be a multiple of 4.
// Out-of-range rows/cols are handled by clamping the load address (their
// products only land in output elements that the guarded store discards).
__global__ __launch_bounds__(32)
void gemm_nt_f32(const float* __restrict__ A, const float* __restrict__ Bt,
                 float* __restrict__ C, int M, int N, int K,
                 const float* __restrict__ bias, int relu,
                 const float* __restrict__ resid) {
    const int lane = threadIdx.x;
    const int hf   = lane >> 4;     // 0/1 -> which K pair / which M half
    const int l16  = lane & 15;
    const int row  = blockIdx.x * 16 + l16;   // A row this lane feeds
    const int coln = blockIdx.y * 16 + l16;   // B row (= output col) this lane feeds
    const int rowc = row  < M ? row  : M - 1; // clamp: keeps loads in-bounds
    const int colc = coln < N ? coln : N - 1;
    const float* __restrict__ Ap = A  + (size_t)rowc * K + 2 * hf;
    const float* __restrict__ Bp = Bt + (size_t)colc * K + 2 * hf;
    __builtin_prefetch(Ap, 0, 1);
    __builtin_prefetch(Bp, 0, 1);
    v8f acc = {};
    for (int k0 = 0; k0 < K; k0 += 4) {
        v2f a, b;
        a.x = Ap[k0];  a.y = Ap[k0 + 1];
        b.x = Bp[k0];  b.y = Bp[k0 + 1];
        acc = __builtin_amdgcn_wmma_f32_16x16x4_f32(false, a, false, b,
                                                    (short)0, acc, false, false);
    }
    const int cc = blockIdx.y * 16 + l16;     // output column this lane stores
    if (cc < N) {
        const float badd = bias ? bias[cc] : 0.f;
        for (int v = 0; v < 8; ++v) {
            const int rr = blockIdx.x * 16 + v + 8 * hf;
            if (rr < M) {
                float val = acc[v] + badd;
                if (relu) val = val > 0.f ? val : 0.f;
                if (resid) val += resid[(size_t)rr * N + cc];
                C[(size_t)rr * N + cc] = val;
            }
        }
    }
}

// -------------------- WMMA GEMM: C = A * B (B stored [K,N]) ----------------
// Used for the inside step: M always a multiple of 16, N multiple of 16.
__global__ __launch_bounds__(32)
void gemm_nn_f32(const float* __restrict__ A, const float* __restrict__ Bm,
                 float* __restrict__ C, int M, int N, int K) {
    const int lane = threadIdx.x;
    const int hf   = lane >> 4;
    const int l16  = lane & 15;
    const int row  = blockIdx.x * 16 + l16;
    const int cn   = blockIdx.y * 16 + l16;
    const float* __restrict__ Ap = A  + (size_t)row * K;
    const float* __restrict__ Bp = Bm + cn;
    __builtin_prefetch(Ap, 0, 1);
    v8f acc = {};
    for (int k0 = 0; k0 < K; k0 += 4) {
        const int ka = k0 + 2 * hf;
        v2f a, b;
        a.x = Ap[ka];
        a.y = Ap[ka + 1];
        b.x = Bp[(size_t)ka * N];
        b.y = Bp[(size_t)(ka + 1) * N];
        acc = __builtin_amdgcn_wmma_f32_16x16x4_f32(false, a, false, b,
                                                    (short)0, acc, false, false);
    }
    for (int v = 0; v < 8; ++v) {
        const int rr = blockIdx.x * 16 + v + 8 * hf;
        C[(size_t)rr * N + cn] = acc[v];
    }
}

// ------------------------- softmax / reductions ----------------------------

// Per-row logsumexp over N elements (one 256-thread block per row).
__global__ void row_lse(const float* __restrict__ logits, float* __restrict__ lse, int N) {
    __shared__ float red[256];
    const int t = blockIdx.x, tid = threadIdx.x;
    const float* row = logits + (size_t)t * N;
    float m = -1e30f;
    for (int i = tid; i < N; i += 256) m = fmaxf(m, row[i]);
    red[tid] = m; __syncthreads();
    for (int s = 128; s; s >>= 1) { if (tid < s) red[tid] = fmaxf(red[tid], red[tid + s]); __syncthreads(); }
    m = red[0]; __syncthreads();
    float sum = 0.f;
    for (int i = tid; i < N; i += 256) sum += __expf(row[i] - m);
    red[tid] = sum; __syncthreads();
    for (int s = 128; s; s >>= 1) { if (tid < s) red[tid] += red[tid + s]; __syncthreads(); }
    if (tid == 0) lse[t] = m + __logf(red[0]);
}

// Per-row (a) softmax over S*S, scattered into W[l*AR_ + a*S_ + r] = erule[a,l,r].
__global__ void rule_softmax_to_W(const float* __restrict__ logits, float* __restrict__ W) {
    __shared__ float red[256];
    const int a = blockIdx.x, tid = threadIdx.x;
    const float* row = logits + (size_t)a * SS_;
    float m = -1e30f;
    for (int i = tid; i < SS_; i += 256) m = fmaxf(m, row[i]);
    red[tid] = m; __syncthreads();
    for (int s = 128; s; s >>= 1) { if (tid < s) red[tid] = fmaxf(red[tid], red[tid + s]); __syncthreads(); }
    m = red[0]; __syncthreads();
    float sum = 0.f;
    for (int i = tid; i < SS_; i += 256) sum += __expf(row[i] - m);
    red[tid] = sum; __syncthreads();
    for (int s = 128; s; s >>= 1) { if (tid < s) red[tid] += red[tid + s]; __syncthreads(); }
    const float inv = 1.f / red[0];
    __syncthreads();
    for (int i = tid; i < SS_; i += 256) {
        const int l = i / S_, r = i % S_;
        W[(size_t)l * AR_ + a * S_ + r] = __expf(row[i] - m) * inv;
    }
}

// log_softmax over NT_ values (single wave).
__global__ __launch_bounds__(32)
void log_softmax_nt(const float* __restrict__ in, float* __restrict__ out) {
    const int lane = threadIdx.x;
    float v = (lane < NT_) ? in[lane] : NEGF;
    float m = v;
    for (int o = 16; o; o >>= 1) m = fmaxf(m, __shfl_xor(m, o, 32));
    float e = (lane < NT_) ? __expf(v - m) : 0.f;
    for (int o = 16; o; o >>= 1) e += __shfl_xor(e, o, 32);
    if (lane < NT_) out[lane] = v - (m + __logf(e));
}

// chart[0][b][s][NT_+t] = logits[t, x[b,s]] - lse[t]
__global__ void unary_fill(const float* __restrict__ logits, const float* __restrict__ lse,
                           const int* __restrict__ x, float* __restrict__ chart) {
    const int idx = blockIdx.x * blockDim.x + threadIdx.x;
    if (idx >= B_ * N_ * T_) return;
    const int t = idx % T_, pos = idx / T_;
    const int b = pos / N_, s = pos % N_;
    const int tok = x[pos];
    chart[(((size_t)0 * B_ + b) * N_ + s) * SP_ + NT_ + t] = logits[(size_t)t * V_ + tok] - lse[t];
}

// ----------------------------- inside step ---------------------------------

// Per row (b,s,u): stabilized exp of left/right span vectors + ml+mr.
__global__ __launch_bounds__(32)
void inside_prep(const float* __restrict__ chart, float* __restrict__ EL,
                 float* __restrict__ ER, float* __restrict__ MLMR, int w) {
    const int ns = N_ - w + 1, wm1 = w - 1;
    const int r = blockIdx.x;
    const int cell = r / wm1, u = (r % wm1) + 1;
    const int b = cell / ns, s = cell % ns;
    const int lane = threadIdx.x;
    const float* L  = chart + (((size_t)(u - 1)     * B_ + b) * N_ + s)     * SP_;
    const float* Rr = chart + (((size_t)(w - u - 1) * B_ + b) * N_ + s + u) * SP_;

    float v0 = (lane < S_)      ? L[lane]      : NEGF;
    float v1 = (lane + 32 < S_) ? L[lane + 32] : NEGF;
    float m = fmaxf(v0, v1);
    for (int o = 16; o; o >>= 1) m = fmaxf(m, __shfl_xor(m, o, 32));
    EL[(size_t)r * KP_ + lane]      = (lane < S_)      ? __expf(v0 - m) : 0.f;
    EL[(size_t)r * KP_ + lane + 32] = (lane + 32 < S_) ? __expf(v1 - m) : 0.f;
    const float ml = m;

    v0 = (lane < S_)      ? Rr[lane]      : NEGF;
    v1 = (lane + 32 < S_) ? Rr[lane + 32] : NEGF;
    m = fmaxf(v0, v1);
    for (int o = 16; o; o >>= 1) m = fmaxf(m, __shfl_xor(m, o, 32));
    ER[(size_t)r * KP_ + lane]      = (lane < S_)      ? __expf(v0 - m) : 0.f;
    ER[(size_t)r * KP_ + lane + 32] = (lane + 32 < S_) ? __expf(v1 - m) : 0.f;
    if (lane == 0) MLMR[r] = ml + m;
}

// v[u,a] = sum_r M1[row,a*S_+r]*ER[row,r]; score = log(v+1e-32)+mlmr;
// chart[w-1][b][s][a] = logsumexp_u(score).  One wave per (b,s) cell.
__global__ __launch_bounds__(32)
void inside_combine(const float* __restrict__ M1, const float* __restrict__ ER,
                    const float* __restrict__ MLMR, float* __restrict__ chart, int w) {
    const int ns = N_ - w + 1, wm1 = w - 1;
    const int cell = blockIdx.x;
    const int b = cell / ns, s = cell % ns;
    const int u = threadIdx.x;                 // split index 0..wm1-1 (lane)
    const int rowbase = cell * wm1;
    const int active = (u < wm1);
    const int row = rowbase + (active ? u : 0);
    const float mlmr = MLMR[row];
    const float* __restrict__ mbase = M1 + (size_t)row * AR_;
    const float* __restrict__ erow  = ER + (size_t)row * KP_;
    float* out = chart + (((size_t)(w - 1) * B_ + b) * N_ + s) * SP_;
    for (int a = 0; a < NT_; ++a) {
        const float* __restrict__ mrow = mbase + a * S_;
        float acc = 0.f;
        for (int r = 0; r < S_; ++r) acc = fmaf(mrow[r], erow[r], acc);
        float sc = active ? (__logf(acc + 1e-32f) + mlmr) : NEGF;
        float mx = sc;
        for (int o = 16; o; o >>= 1) mx = fmaxf(mx, __shfl_xor(mx, o, 32));
        float ex = active ? __expf(sc - mx) : 0.f;
        for (int o = 16; o; o >>= 1) ex += __shfl_xor(ex, o, 32);
        if (u == 0) out[a] = mx + __logf(ex);
    }
}

// out[b] = -logsumexp_a(chart[n-1][b][0][a] + rootlp[a])
__global__ __launch_bounds__(32)
void final_nll(const float* __restrict__ chart, const float* __restrict__ rootlp,
               float* __restrict__ out) {
    const int b = blockIdx.x, lane = threadIdx.x;
    const float* c = chart + (((size_t)(N_ - 1) * B_ + b) * N_ + 0) * SP_;
    float v = (lane < NT_) ? c[lane] + rootlp[lane] : NEGF;
    float m = v;
    for (int o = 16; o; o >>= 1) m = fmaxf(m, __shfl_xor(m, o, 32));
    float e = (lane < NT_) ? __expf(v - m) : 0.f;
    for (int o = 16; o; o >>= 1) e += __shfl_xor(e, o, 32);
    if (lane == 0) out[b] = -(m + __logf(e));
}

// ------------------------------ launcher -----------------------------------

extern "C" void kernel_launch(void* const* d_in, const int* in_sizes, int n_in,
                              void* d_out, int out_size, void* d_ws, size_t ws_size,
                              hipStream_t stream) {
    (void)in_sizes; (void)n_in; (void)out_size; (void)ws_size;
    const int*   x        = (const int*)  d_in[0];
    const float* t_emb    = (const float*)d_in[1];
    const float* nt_emb   = (const float*)d_in[2];
    const float* root_emb = (const float*)d_in[3];
    const float* rule_W   = (const float*)d_in[4];
    const float* rule_b   = (const float*)d_in[5];
    const float* root_W0  = (const float*)d_in[6];
    const float* root_b0  = (const float*)d_in[7];
    const float* root_rW  = (const float*)d_in[8];   // [4,256,256]
    const float* root_rb  = (const float*)d_in[9];   // [4,256]
    const float* root_Wo  = (const float*)d_in[10];
    const float* root_bo  = (const float*)d_in[11];
    const float* voc_W0   = (const float*)d_in[12];
    const float* voc_b0   = (const float*)d_in[13];
    const float* voc_rW   = (const float*)d_in[14];
    const float* voc_rb   = (const float*)d_in[15];
    const float* voc_Wo   = (const float*)d_in[16];
    const float* voc_bo   = (const float*)d_in[17];
    float* out = (float*)d_out;

    // ---- workspace layout (floats) ----
    float* ws = (float*)d_ws;
    size_t o = 0;
    float* chart   = ws + o; o += (size_t)N_ * B_ * N_ * SP_;   // 589824
    float* logits  = ws + o; o += (size_t)T_ * V_;              // 400000
    float* lse     = ws + o; o += 64;
    float* hA      = ws + o; o += T_ * D_;
    float* hB      = ws + o; o += T_ * D_;
    float* hC      = ws + o; o += T_ * D_;
    float* rh      = ws + o; o += D_;
    float* rt1     = ws + o; o += D_;
    float* rt2     = ws + o; o += D_;
    float* rlog    = ws + o; o += 32;
    float* rlp     = ws + o; o += 32;
    float* rulelog = ws + o; o += NT_ * SS_;                    // 72000
    float* Wi      = ws + o; o += (size_t)KP_ * AR_;            // 76800
    float* EL      = ws + o; o += (size_t)RMAX_ * KP_;
    float* ER      = ws + o; o += (size_t)RMAX_ * KP_;
    float* MLMR    = ws + o; o += RMAX_;
    float* M1      = ws + o; o += (size_t)RMAX_ * AR_;          // 2764800

    auto g16 = [](int m) { return (m + 15) / 16; };
    const dim3 wv(32);

    // ---- init ----
    {
        size_t cc = (size_t)N_ * B_ * N_ * SP_;
        fill_f32<<<dim3((unsigned)((cc + 255) / 256)), 256, 0, stream>>>(chart, NEGF, cc);
        size_t wc = (size_t)KP_ * AR_;
        fill_f32<<<dim3((unsigned)((wc + 255) / 256)), 256, 0, stream>>>(Wi, 0.f, wc);
    }

    // ---- root MLP -> rlp[NT] ----
    gemm_nt_f32<<<dim3(1, 16), wv, 0, stream>>>(root_emb, root_W0, rh, 1, D_, D_, root_b0, 0, nullptr);
    gemm_nt_f32<<<dim3(1, 16), wv, 0, stream>>>(rh,  root_rW + 0 * D_ * D_, rt1, 1, D_, D_, root_rb + 0 * D_, 1, nullptr);
    gemm_nt_f32<<<dim3(1, 16), wv, 0, stream>>>(rt1, root_rW + 1 * D_ * D_, rt2, 1, D_, D_, root_rb + 1 * D_, 1, rh);
    gemm_nt_f32<<<dim3(1, 16), wv, 0, stream>>>(rt2, root_rW + 2 * D_ * D_, rt1, 1, D_, D_, root_rb + 2 * D_, 1, nullptr);
    gemm_nt_f32<<<dim3(1, 16), wv, 0, stream>>>(rt1, root_rW + 3 * D_ * D_, rh,  1, D_, D_, root_rb + 3 * D_, 1, rt2);
    gemm_nt_f32<<<dim3(1, g16(NT_)), wv, 0, stream>>>(rh, root_Wo, rlog, 1, NT_, D_, root_bo, 0, nullptr);
    log_softmax_nt<<<1, wv, 0, stream>>>(rlog, rlp);

    // ---- vocab MLP -> emission logits + per-terminal lse ----
    gemm_nt_f32<<<dim3(g16(T_), 16), wv, 0, stream>>>(t_emb, voc_W0, hA, T_, D_, D_, voc_b0, 0, nullptr);
    gemm_nt_f32<<<dim3(g16(T_), 16), wv, 0, stream>>>(hA, voc_rW + 0 * D_ * D_, hB, T_, D_, D_, voc_rb + 0 * D_, 1, nullptr);
    gemm_nt_f32<<<dim3(g16(T_), 16), wv, 0, stream>>>(hB, voc_rW + 1 * D_ * D_, hC, T_, D_, D_, voc_rb + 1 * D_, 1, hA);
    gemm_nt_f32<<<dim3(g16(T_), 16), wv, 0, stream>>>(hC, voc_rW + 2 * D_ * D_, hB, T_, D_, D_, voc_rb + 2 * D_, 1, nullptr);
    gemm_nt_f32<<<dim3(g16(T_), 16), wv, 0, stream>>>(hB, voc_rW + 3 * D_ * D_, hA, T_, D_, D_, voc_rb + 3 * D_, 1, hC);
    gemm_nt_f32<<<dim3(g16(T_), V_ / 16), wv, 0, stream>>>(hA, voc_Wo, logits, T_, V_, D_, voc_bo, 0, nullptr);
    row_lse<<<T_, 256, 0, stream>>>(logits, lse, V_);
    unary_fill<<<(B_ * N_ * T_ + 255) / 256, 256, 0, stream>>>(logits, lse, x, chart);

    // ---- rule scores -> Wi[l, a*S+r] = softmax_a(rule)[a,l,r] ----
    gemm_nt_f32<<<dim3(g16(NT_), SS_ / 16), wv, 0, stream>>>(nt_emb, rule_W, rulelog, NT_, SS_, D_, rule_b, 0, nullptr);
    rule_softmax_to_W<<<NT_, 256, 0, stream>>>(rulelog, Wi);

    // ---- inside algorithm: widths 2..N ----
    for (int w = 2; w <= N_; ++w) {
        const int ns = N_ - w + 1;
        const int cells = B_ * ns;
        const int R = cells * (w - 1);       // always a multiple of 16 (B_=16)
        inside_prep<<<R, wv, 0, stream>>>(chart, EL, ER, MLMR, w);
        gemm_nn_f32<<<dim3(R / 16, AR_ / 16), wv, 0, stream>>>(EL, Wi, M1, R, AR_, KP_);
        inside_combine<<<cells, wv, 0, stream>>>(M1, ER, MLMR, chart, w);
    }

    // ---- final NLL ----
    final_nll<<<B_, wv, 0, stream>>>(chart, rlp, out);
}